// ARModel_51058571214850
// MI455X (gfx1250) — compile-verified
//
#include <hip/hip_runtime.h>
#include <hip/hip_bf16.h>

typedef __attribute__((ext_vector_type(2))) float v2f;
typedef __attribute__((ext_vector_type(8))) float v8f;

#define BATCH   16384
#define NLAG    64
#define STEPS   4096
#define TBLK    64
#define NBLK    (STEPS / TBLK)   // 64 serial block-rounds
#define LSTR    68               // padded LDS row stride (floats) -> bank-conflict-free

#define WAVES_PER_WG   4
#define ROWS_PER_WAVE  16
#define ROWS_PER_WG    (WAVES_PER_WG * ROWS_PER_WAVE)

// ---------------------------------------------------------------------------
// Kernel 1: build block-recurrence matrices.
//   x_{t+j} = sum_i Wa[i][j] * s[i] + sum_m (sigma*Wb[m][j]) * eps_m
// Stored transposed for direct B-fragment b64 loads:
//   mats[0..4095]    = Wa^T[j][i]        (j = output step in block, i = state idx)
//   mats[4096..8191] = sigma * Wb^T[j][m]
// Each thread owns one column d of the 128-dim coefficient vectors (64 state
// dims + 64 noise dims) -> no synchronization needed in the recurrence loop.
// ---------------------------------------------------------------------------
__global__ __launch_bounds__(128) void ar_prep(const float* __restrict__ coef,
                                               const float* __restrict__ log_noise,
                                               float* __restrict__ mats) {
  __shared__ float V[64 * 128];   // V[j][d] = d-th component of v_j
  __shared__ float c[64];
  const int d = threadIdx.x;      // 0..127
  if (d < 64) c[d] = coef[d];
  __syncthreads();

  for (int j = 0; j < 64; ++j) {
    // identity (pre-block state) contributions: lag i with j+i < 64
    float val = (d < 64 && d >= j) ? c[d - j] : 0.0f;
    if (d == 64 + j) val += 1.0f;              // direct eps_j term
    // contributions from already-computed in-block outputs v_q (q < j)
    for (int q = 0; q < j; ++q)
      val += c[64 - j + q] * V[q * 128 + d];
    V[j * 128 + d] = val;                      // thread-private column: no sync
  }
  __syncthreads();

  const float sigma = __expf(log_noise[0]);
  for (int idx = d; idx < 64 * 64; idx += 128) {
    const int j = idx >> 6;
    const int i = idx & 63;
    mats[idx]        = V[j * 128 + i];
    mats[4096 + idx] = sigma * V[j * 128 + 64 + i];
  }
}

// ---------------------------------------------------------------------------
// Kernel 2: 64 rounds of  Y[16x64] = S[16x64]@Wa + noise[16x64]@(sigma*Wb)
// per wave (16 batch rows), using V_WMMA_F32_16X16X4_F32.
//   A-frag (16x4 f32): lane L holds row M=L%16, K = 2*(L/16)+v  -> one b64 load
//   B-frag (4x16 f32): lane L holds col N=L%16, K = 2*(L/16)+v  -> ds_load_b64
//   D      (16x16)   : VGPR r, lane L -> M = r+8*(L/16), N = L%16
// LDS stride 68 floats makes every fragment access hit 64 distinct banks.
// ---------------------------------------------------------------------------
__global__ __launch_bounds__(128) void ar_main(const float* __restrict__ init,
                                               const float* __restrict__ noise,
                                               const float* __restrict__ mats,
                                               float* __restrict__ out) {
  __shared__ float lds[(128 + ROWS_PER_WG) * LSTR];   // Wa(64) + Wb(64) + state(4x16) rows
  float* Wa = lds;
  float* Wb = lds + 64 * LSTR;

  const int tid  = threadIdx.x;
  const int wave = tid >> 5;
  const int lane = tid & 31;
  const int lm   = lane & 15;   // row / col within tile
  const int lg   = lane >> 4;   // lane-group (K half)
  float* St = lds + 128 * LSTR + wave * ROWS_PER_WAVE * LSTR;

  // stage Wa / Wb into padded LDS
  for (int idx = tid; idx < 64 * 64; idx += 128) {
    const int j = idx >> 6, k = idx & 63;
    Wa[j * LSTR + k] = mats[idx];
    Wb[j * LSTR + k] = mats[4096 + idx];
  }
  __syncthreads();

  const int r0 = (blockIdx.x * WAVES_PER_WG + wave) * ROWS_PER_WAVE;
  const float* ivrow = init  + (size_t)(r0 + lm) * NLAG;
  const float* nzrow = noise + (size_t)(r0 + lm) * STEPS;

  // initial state A-fragments (state = initial window, oldest first)
  v2f sA[16];
#pragma unroll
  for (int ks = 0; ks < 16; ++ks)
    sA[ks] = *(const v2f*)(ivrow + 4 * ks + 2 * lg);

  for (int b = 0; b < NBLK; ++b) {
    const int t0 = b * TBLK;

    // issue noise loads early; their latency hides behind the state-path WMMAs
    v2f nz[16];
#pragma unroll
    for (int ks = 0; ks < 16; ++ks)
      nz[ks] = *(const v2f*)(nzrow + t0 + 4 * ks + 2 * lg);

    v8f acc0 = {}, acc1 = {}, acc2 = {}, acc3 = {};

    // state path: acc += S @ Wa
#pragma unroll
    for (int ks = 0; ks < 16; ++ks) {
      const int kb = 4 * ks + 2 * lg;
      const float* w = Wa + lm * LSTR + kb;
      const v2f w0 = *(const v2f*)(w);
      const v2f w1 = *(const v2f*)(w + 16 * LSTR);
      const v2f w2 = *(const v2f*)(w + 32 * LSTR);
      const v2f w3 = *(const v2f*)(w + 48 * LSTR);
      acc0 = __builtin_amdgcn_wmma_f32_16x16x4_f32(false, sA[ks], false, w0, (short)0, acc0, false, false);
      acc1 = __builtin_amdgcn_wmma_f32_16x16x4_f32(false, sA[ks], false, w1, (short)0, acc1, false, false);
      acc2 = __builtin_amdgcn_wmma_f32_16x16x4_f32(false, sA[ks], false, w2, (short)0, acc2, false, false);
      acc3 = __builtin_amdgcn_wmma_f32_16x16x4_f32(false, sA[ks], false, w3, (short)0, acc3, false, false);
    }
    // noise path: acc += noise @ (sigma*Wb)
#pragma unroll
    for (int ks = 0; ks < 16; ++ks) {
      const int kb = 4 * ks + 2 * lg;
      const float* w = Wb + lm * LSTR + kb;
      const v2f w0 = *(const v2f*)(w);
      const v2f w1 = *(const v2f*)(w + 16 * LSTR);
      const v2f w2 = *(const v2f*)(w + 32 * LSTR);
      const v2f w3 = *(const v2f*)(w + 48 * LSTR);
      acc0 = __builtin_amdgcn_wmma_f32_16x16x4_f32(false, nz[ks], false, w0, (short)0, acc0, false, false);
      acc1 = __builtin_amdgcn_wmma_f32_16x16x4_f32(false, nz[ks], false, w1, (short)0, acc1, false, false);
      acc2 = __builtin_amdgcn_wmma_f32_16x16x4_f32(false, nz[ks], false, w2, (short)0, acc2, false, false);
      acc3 = __builtin_amdgcn_wmma_f32_16x16x4_f32(false, nz[ks], false, w3, (short)0, acc3, false, false);
    }

    // write block outputs + stage new state ([m][k] layout) in LDS
#pragma unroll
    for (int r = 0; r < 8; ++r) {
      float* orow = out + (size_t)(r0 + r + 8 * lg) * STEPS + t0 + lm;
      orow[0]  = acc0[r];
      orow[16] = acc1[r];
      orow[32] = acc2[r];
      orow[48] = acc3[r];
      float* srow = St + (r + 8 * lg) * LSTR + lm;
      srow[0]  = acc0[r];
      srow[16] = acc1[r];
      srow[32] = acc2[r];
      srow[48] = acc3[r];
    }
    // same-wave LDS ops execute in order; the wait also fences compiler motion
    asm volatile("s_wait_dscnt 0" ::: "memory");
#pragma unroll
    for (int ks = 0; ks < 16; ++ks)
      sA[ks] = *(const v2f*)(St + lm * LSTR + 4 * ks + 2 * lg);
    asm volatile("s_wait_dscnt 0" ::: "memory");
  }
}

extern "C" void kernel_launch(void* const* d_in, const int* in_sizes, int n_in,
                              void* d_out, int out_size, void* d_ws, size_t ws_size,
                              hipStream_t stream) {
  const float* init  = (const float*)d_in[0];   // [16384, 64]
  const float* coef  = (const float*)d_in[1];   // [64]
  const float* lns   = (const float*)d_in[2];   // scalar
  const float* noise = (const float*)d_in[3];   // [16384, 4096]
  float* mats = (float*)d_ws;                   // 2*64*64 floats = 32 KB scratch

  ar_prep<<<1, 128, 0, stream>>>(coef, lns, mats);
  ar_main<<<BATCH / ROWS_PER_WG, 128, 0, stream>>>(init, noise, mats, (float*)d_out);
}